// RelationalKENN_37254546326227
// MI455X (gfx1250) — compile-verified
//
#include <hip/hip_runtime.h>
#include <math.h>

#define N_NODES  100000
#define N_EDGES  1600000
#define N_UNARY  16
#define N_BINARY 4

#define LOG2E 1.44269504088896340736f

// Native-exp sigmoid: 1/(1+exp(-t)) via v_exp_f32 (exp2) + v_rcp_f32.
// Saturation: t->+inf => rcp(1+0)=1 ; t->-inf => rcp(inf)=0.  ~2 ulp.
__device__ __forceinline__ float fast_exp(float x) {
    return __builtin_amdgcn_exp2f(x * LOG2E);
}
__device__ __forceinline__ float fast_rcp(float x) {
    return __builtin_amdgcn_rcpf(x);
}
__device__ __forceinline__ float sigmoidf_(float t) {
    return fast_rcp(1.0f + fast_exp(-t));
}

__device__ __forceinline__ void softmax3(float a, float b, float c,
                                         float& s0, float& s1, float& s2) {
    float m  = fmaxf(a, fmaxf(b, c));      // v_max3_num_f32
    float e0 = fast_exp(a - m);            // args <= 0: no overflow possible
    float e1 = fast_exp(b - m);
    float e2 = fast_exp(c - m);
    float inv = fast_rcp(e0 + e1 + e2);
    s0 = e0 * inv; s1 = e1 * inv; s2 = e2 * inv;
}

// ---------------------------------------------------------------------------
// Kernel 1: u = unary + unary_enhancer(unary).  One thread per node.
// Clauses: ([0,1],[-1,1]) ([2,3],[-1,1]) ([4,5,6],[-1,1,1]) ([7,8],[1,-1])
// ---------------------------------------------------------------------------
__global__ void kenn_unary(const float* __restrict__ unary,
                           const float* __restrict__ ucw,
                           float* __restrict__ out_u) {
    int i = blockIdx.x * blockDim.x + threadIdx.x;
    if (i >= N_NODES) return;

    const float4* row = (const float4*)(unary + (size_t)i * N_UNARY);
    float4 r0 = row[0];   // x0..x3
    float4 r1 = row[1];   // x4..x7
    float4 r2 = row[2];   // x8..x11
    float4 r3 = row[3];   // x12..x15

    float w0 = ucw[0], w1 = ucw[1], w2 = ucw[2], w3 = ucw[3];

    // clause 0: sel=[-x0, x1] -> sm0=sig(-(x0+x1)), sm1=sig(x0+x1)
    float t01 = r0.x + r0.y;
    r0.x -= sigmoidf_(-t01) * w0;
    r0.y += sigmoidf_( t01) * w0;

    // clause 1: sel=[-x2, x3]
    float t23 = r0.z + r0.w;
    r0.z -= sigmoidf_(-t23) * w1;
    r0.w += sigmoidf_( t23) * w1;

    // clause 2: sel=[-x4, x5, x6], signs [-1, 1, 1]
    float s0, s1, s2;
    softmax3(-r1.x, r1.y, r1.z, s0, s1, s2);
    r1.x -= s0 * w2;
    r1.y += s1 * w2;
    r1.z += s2 * w2;

    // clause 3: sel=[x7, -x8], signs [1, -1]
    //   sm0 = sig(x7+x8), sm1 = sig(-(x7+x8));  d7=+sm0*w, d8=-sm1*w
    float t78 = r1.w + r2.x;
    r1.w += sigmoidf_( t78) * w3;
    r2.x -= sigmoidf_(-t78) * w3;

    float4* orow = (float4*)(out_u + (size_t)i * N_UNARY);
    orow[0] = r0; orow[1] = r1; orow[2] = r2; orow[3] = r3;
}

// ---------------------------------------------------------------------------
// Kernel 2: per-edge binary clauses + segment-sum scatter via f32 atomics.
// joined layout: [0:16)=u[i1], [16:32)=u[i2], [32:36)=binary.
// Clauses: ([0,32,16],[-1,-1,1]) ([1,33,17],[-1,-1,1])
//          ([34,35],[-1,1])      ([2,18],[-1,1])
// Only u columns 0..2 are needed -> recompute from raw unary cols 0..3
// (clean read, independent of the atomics landing in out_u).
// ---------------------------------------------------------------------------
__global__ void kenn_binary(const float* __restrict__ unary,
                            const float* __restrict__ binary,
                            const int* __restrict__ index1,
                            const int* __restrict__ index2,
                            const float* __restrict__ ucw,
                            const float* __restrict__ bcw,
                            float* __restrict__ out_u,
                            float* __restrict__ out_b) {
    int e = blockIdx.x * blockDim.x + threadIdx.x;
    if (e >= N_EDGES) return;

    int i1 = index1[e];
    int i2 = index2[e];
    const float* p1 = unary + (size_t)i1 * N_UNARY;
    const float* p2 = unary + (size_t)i2 * N_UNARY;
    // gfx1250 global_prefetch_b8: kick the two random L2 gathers early
    __builtin_prefetch(p1, 0, 0);
    __builtin_prefetch(p2, 0, 0);

    float4 bin = ((const float4*)binary)[e];     // streamed, coalesced b128
    float4 xa  = *(const float4*)p1;             // unary[i1][0..3]
    float4 xb  = *(const float4*)p2;             // unary[i2][0..3]

    float w0 = ucw[0], w1 = ucw[1];
    float bw0 = bcw[0], bw1 = bcw[1], bw2 = bcw[2], bw3 = bcw[3];

    // recompute u[i1][0..2]
    float tA  = xa.x + xa.y;
    float a0  = xa.x - sigmoidf_(-tA)  * w0;
    float a1  = xa.y + sigmoidf_( tA)  * w0;
    float tA2 = xa.z + xa.w;
    float a2  = xa.z - sigmoidf_(-tA2) * w1;

    // recompute u[i2][0..2]
    float tB  = xb.x + xb.y;
    float b0  = xb.x - sigmoidf_(-tB)  * w0;
    float b1  = xb.y + sigmoidf_( tB)  * w0;
    float tB2 = xb.z + xb.w;
    float b2  = xb.z - sigmoidf_(-tB2) * w1;

    float s0, s1, s2;

    // clause B0: sel = [-a0, -bin0, b0]
    softmax3(-a0, -bin.x, b0, s0, s1, s2);
    float d0  = -s0 * bw0;
    float db0 = -s1 * bw0;
    float d16 =  s2 * bw0;

    // clause B1: sel = [-a1, -bin1, b1]
    softmax3(-a1, -bin.y, b1, s0, s1, s2);
    float d1  = -s0 * bw1;
    float db1 = -s1 * bw1;
    float d17 =  s2 * bw1;

    // clause B2: sel = [-bin2, bin3]
    float tC  = bin.z + bin.w;
    float db2 = -sigmoidf_(-tC) * bw2;
    float db3 =  sigmoidf_( tC) * bw2;

    // clause B3: sel = [-a2, b2]
    float tD  = a2 + b2;
    float d2  = -sigmoidf_(-tD) * bw3;
    float d18 =  sigmoidf_( tD) * bw3;

    // binary output: binary + delta_bp   (coalesced b128 store)
    ((float4*)out_b)[e] = make_float4(bin.x + db0, bin.y + db1,
                                      bin.z + db2, bin.w + db3);

    // segment_sum scatter: 6 non-returning global_atomic_add_f32
    float* o1 = out_u + (size_t)i1 * N_UNARY;
    float* o2 = out_u + (size_t)i2 * N_UNARY;
    atomicAdd(o1 + 0, d0);
    atomicAdd(o1 + 1, d1);
    atomicAdd(o1 + 2, d2);
    atomicAdd(o2 + 0, d16);
    atomicAdd(o2 + 1, d17);
    atomicAdd(o2 + 2, d18);
}

extern "C" void kernel_launch(void* const* d_in, const int* in_sizes, int n_in,
                              void* d_out, int out_size, void* d_ws, size_t ws_size,
                              hipStream_t stream) {
    const float* unary  = (const float*)d_in[0];
    const float* binary = (const float*)d_in[1];
    const int*   index1 = (const int*)d_in[2];
    const int*   index2 = (const int*)d_in[3];
    const float* ucw    = (const float*)d_in[4];
    const float* bcw    = (const float*)d_in[5];

    float* out_u = (float*)d_out;                       // N_NODES * N_UNARY
    float* out_b = out_u + (size_t)N_NODES * N_UNARY;   // N_EDGES * N_BINARY

    (void)in_sizes; (void)n_in; (void)out_size; (void)d_ws; (void)ws_size;

    // Kernel 1 fully rewrites out_u = u each call (no stale accumulation),
    // then kernel 2 (same stream, ordered) accumulates delta_up atomically.
    kenn_unary<<<(N_NODES + 255) / 256, 256, 0, stream>>>(unary, ucw, out_u);
    kenn_binary<<<(N_EDGES + 255) / 256, 256, 0, stream>>>(
        unary, binary, index1, index2, ucw, bcw, out_u, out_b);
}